// three_gcn_10033043604178
// MI455X (gfx1250) — compile-verified
//
#include <hip/hip_runtime.h>
#include <hip/hip_bf16.h>
#include <math.h>

// ---------------------------------------------------------------------------
// 3-layer GCN for MI455X (gfx1250, wave32).
//   y_l = act( dst_norm ⊙ scatter_add(src_norm ⊙ feat)[edges] @ W_l + b_l )
// SpMM is the bandwidth term (~5 GB total -> ~0.22 ms @ 23.3 TB/s); GEMM is
// ~10 GFLOP total, done with V_WMMA_F32_16X16X4_F32 (fp32 matrix path, keeps
// reference precision). A tiles are staged global->LDS with the CDNA5 async
// LDS path (ASYNCcnt) when the toolchain exposes the builtins.
// ---------------------------------------------------------------------------

typedef float v2f __attribute__((ext_vector_type(2)));
typedef float v8f __attribute__((ext_vector_type(8)));
typedef int   v4i __attribute__((ext_vector_type(4)));

#define D 128                 // feature dim (in == out == 128)
#define LDS_STRIDE 132        // 128+4 pad: k-broadcast reads hit 16 distinct banks
                              // 132 floats = 528 B = 33*16 -> rows stay 16B-aligned

#if __has_builtin(__builtin_amdgcn_global_load_async_to_lds_b128) && \
    __has_builtin(__builtin_amdgcn_s_wait_asynccnt)
#define USE_ASYNC_LDS 1
typedef __attribute__((address_space(1))) v4i* g_v4i_p;   // global 16B chunk
typedef __attribute__((address_space(3))) v4i* l_v4i_p;   // LDS 16B chunk
#else
#define USE_ASYNC_LDS 0
#endif

// ---- degrees via float atomics (exact for counts < 2^24) -------------------
__global__ void degree_kernel(const int* __restrict__ esrc,
                              const int* __restrict__ edst,
                              float* __restrict__ deg_src,
                              float* __restrict__ deg_dst,
                              int n_edges) {
  int e = blockIdx.x * blockDim.x + threadIdx.x;
  if (e >= n_edges) return;
  atomicAdd(&deg_src[esrc[e]], 1.0f);
  atomicAdd(&deg_dst[edst[e]], 1.0f);
}

// ---- norm = rsqrt(max(deg,1)) in place over both arrays (2N elems) ---------
__global__ void norm_kernel(float* __restrict__ deg, int n2) {
  int i = blockIdx.x * blockDim.x + threadIdx.x;
  if (i >= n2) return;
  deg[i] = rsqrtf(fmaxf(deg[i], 1.0f));
}

// ---- SpMM scatter: one wave32 per edge, 4 floats/lane ----------------------
// Coalesced 512B row gather (scaled by src_norm) + 4 global_atomic_add_f32
// per lane into the destination row. Accumulator (52MB) is L2-resident, so
// the atomics resolve at the L2 atomic units, not HBM.
__global__ void scatter_kernel(const float* __restrict__ feat,
                               const float* __restrict__ src_norm,
                               const int* __restrict__ esrc,
                               const int* __restrict__ edst,
                               float* __restrict__ agg,
                               int n_edges) {
  int gwave = (blockIdx.x * blockDim.x + threadIdx.x) >> 5;  // edge id
  int lane  = threadIdx.x & 31;
  if (gwave >= n_edges) return;
  int s = esrc[gwave];
  int d = edst[gwave];
  float sn = src_norm[s];
  const float4 v = *(const float4*)(feat + s * D + lane * 4);
  float* dst = agg + d * D + lane * 4;
  atomicAdd(dst + 0, v.x * sn);
  atomicAdd(dst + 1, v.y * sn);
  atomicAdd(dst + 2, v.z * sn);
  atomicAdd(dst + 3, v.w * sn);
}

// ---- GEMM: out[16 rows x 128 cols] per block; 8 waves x one 16x16 tile -----
// A-frag packing per ISA 7.12.2 (32-bit A 16x4): lanes 0-15 hold K=k0,k0+1
// for row M=lane; lanes 16-31 hold K=k0+2,k0+3 for row M=lane-16. B-frag
// mirrors (lanes 0-15: rows K=k0,k0+1 at col N=lane; lanes 16-31: K=k0+2,3).
// C/D: VGPR j -> M=j (lanes 0-15) / M=j+8 (lanes 16-31), N=lane&15.
// dst_norm is applied in the epilogue: (diag(dn)*A)*W == diag(dn)*(A*W),
// which makes the A staging a pure copy -> async global->LDS (B128, ASYNCcnt).
// act: 0 = identity (layer 3 logits), 1 = ELU.
__global__ __launch_bounds__(256)
void gemm_wmma_kernel(const float* __restrict__ agg,
                      const float* __restrict__ dst_norm,
                      const float* __restrict__ W,   // [128 x 128], row-major [k][n]
                      const float* __restrict__ bias,
                      float* __restrict__ out,
                      int act) {
  __shared__ float As[16 * LDS_STRIDE];

  const int row0 = blockIdx.x * 16;
  const int tid  = threadIdx.x;

#if USE_ASYNC_LDS
  {
    // 16x128 tile = 2048 floats; each thread moves two 16B chunks.
    const float* g = agg + row0 * D;
    int i0 = tid * 4;                 // 0..1020, 16B-aligned chunks
    int i1 = i0 + 1024;
    int r0 = i0 >> 7, c0 = i0 & 127;
    int r1 = i1 >> 7, c1 = i1 & 127;
    __builtin_amdgcn_global_load_async_to_lds_b128(
        (g_v4i_p)((void*)(g + i0)),
        (l_v4i_p)((void*)&As[r0 * LDS_STRIDE + c0]),
        0, 0);
    __builtin_amdgcn_global_load_async_to_lds_b128(
        (g_v4i_p)((void*)(g + i1)),
        (l_v4i_p)((void*)&As[r1 * LDS_STRIDE + c1]),
        0, 0);
    __builtin_amdgcn_s_wait_asynccnt(0);
  }
#else
  {
    int r = tid >> 4;            // 0..15
    int c = (tid & 15) * 8;      // 0,8,...,120
    const float4 v0 = *(const float4*)(agg + (row0 + r) * D + c);
    const float4 v1 = *(const float4*)(agg + (row0 + r) * D + c + 4);
    *(float4*)&As[r * LDS_STRIDE + c]     = v0;
    *(float4*)&As[r * LDS_STRIDE + c + 4] = v1;
  }
#endif
  __syncthreads();

  const int wave = tid >> 5;        // 0..7 -> column tile
  const int lane = tid & 31;
  const int half = lane >> 4;       // 0: K pair {0,1}; 1: K pair {2,3}
  const int l    = lane & 15;       // row (A) / col (B) within tile
  const int n0   = wave * 16;

  v8f c = {0.f, 0.f, 0.f, 0.f, 0.f, 0.f, 0.f, 0.f};

#pragma unroll
  for (int k0 = 0; k0 < D; k0 += 4) {
    const int ka = k0 + 2 * half;
    v2f a, b;
    a.x = As[l * LDS_STRIDE + ka];
    a.y = As[l * LDS_STRIDE + ka + 1];
    b.x = W[ka * D + n0 + l];
    b.y = W[(ka + 1) * D + n0 + l];
    // (neg_a, A, neg_b, B, c_mod, C, reuse_a, reuse_b)
    c = __builtin_amdgcn_wmma_f32_16x16x4_f32(false, a, false, b,
                                              (short)0, c, false, false);
  }

  const float bv = bias[n0 + l];
#pragma unroll
  for (int j = 0; j < 8; ++j) {
    const int m = j + 8 * half;
    float v = c[j] * dst_norm[row0 + m] + bv;   // fused dst_norm row scale
    if (act) v = (v > 0.0f) ? v : expm1f(v);    // ELU(alpha=1)
    out[(row0 + m) * D + n0 + l] = v;
  }
}

// ---- row softmax over 128 cols, one wave32 per node, in place --------------
__global__ void softmax_kernel(float* __restrict__ y, int n_nodes) {
  int node = (blockIdx.x * blockDim.x + threadIdx.x) >> 5;
  int lane = threadIdx.x & 31;
  if (node >= n_nodes) return;
  float4 v = *(const float4*)(y + node * D + lane * 4);
  float m = fmaxf(fmaxf(v.x, v.y), fmaxf(v.z, v.w));
#pragma unroll
  for (int off = 16; off; off >>= 1) m = fmaxf(m, __shfl_xor(m, off, 32));
  float4 e;
  e.x = __expf(v.x - m); e.y = __expf(v.y - m);
  e.z = __expf(v.z - m); e.w = __expf(v.w - m);
  float s = e.x + e.y + e.z + e.w;
#pragma unroll
  for (int off = 16; off; off >>= 1) s += __shfl_xor(s, off, 32);
  float inv = 1.0f / s;
  e.x *= inv; e.y *= inv; e.z *= inv; e.w *= inv;
  *(float4*)(y + node * D + lane * 4) = e;
}

extern "C" void kernel_launch(void* const* d_in, const int* in_sizes, int n_in,
                              void* d_out, int out_size, void* d_ws, size_t ws_size,
                              hipStream_t stream) {
  const float* x  = (const float*)d_in[0];
  const float* W1 = (const float*)d_in[1];
  const float* b1 = (const float*)d_in[2];
  const float* W2 = (const float*)d_in[3];
  const float* b2 = (const float*)d_in[4];
  const float* W3 = (const float*)d_in[5];
  const float* b3 = (const float*)d_in[6];
  const int* esrc = (const int*)d_in[7];
  const int* edst = (const int*)d_in[8];

  const int n_nodes = in_sizes[0] / D;   // 102400
  const int n_edges = in_sizes[7];       // 1638400

  // Workspace layout (floats): deg/src_norm [N] | deg/dst_norm [N] | agg [N*D]
  float* ws       = (float*)d_ws;
  float* src_norm = ws;
  float* dst_norm = ws + n_nodes;
  float* agg      = ws + 2 * n_nodes;

  float* y1 = (float*)d_out;
  float* y2 = y1 + (size_t)n_nodes * D;
  float* y3 = y2 + (size_t)n_nodes * D;

  // Degrees -> norms.
  (void)hipMemsetAsync(src_norm, 0, (size_t)2 * n_nodes * sizeof(float), stream);
  degree_kernel<<<(n_edges + 255) / 256, 256, 0, stream>>>(
      esrc, edst, src_norm, dst_norm, n_edges);
  norm_kernel<<<(2 * n_nodes + 255) / 256, 256, 0, stream>>>(src_norm, 2 * n_nodes);

  const int scatter_blocks = (n_edges + 7) / 8;        // 8 edges (waves) / block
  const int gemm_blocks    = n_nodes / 16;             // 16-row tiles

  // Layer 1: x -> y1 (ELU)
  (void)hipMemsetAsync(agg, 0, (size_t)n_nodes * D * sizeof(float), stream);
  scatter_kernel<<<scatter_blocks, 256, 0, stream>>>(x, src_norm, esrc, edst, agg, n_edges);
  gemm_wmma_kernel<<<gemm_blocks, 256, 0, stream>>>(agg, dst_norm, W1, b1, y1, 1);

  // Layer 2: y1 -> y2 (ELU)
  (void)hipMemsetAsync(agg, 0, (size_t)n_nodes * D * sizeof(float), stream);
  scatter_kernel<<<scatter_blocks, 256, 0, stream>>>(y1, src_norm, esrc, edst, agg, n_edges);
  gemm_wmma_kernel<<<gemm_blocks, 256, 0, stream>>>(agg, dst_norm, W2, b2, y2, 1);

  // Layer 3: y2 -> y3 logits, then row softmax in place.
  (void)hipMemsetAsync(agg, 0, (size_t)n_nodes * D * sizeof(float), stream);
  scatter_kernel<<<scatter_blocks, 256, 0, stream>>>(y2, src_norm, esrc, edst, agg, n_edges);
  gemm_wmma_kernel<<<gemm_blocks, 256, 0, stream>>>(agg, dst_norm, W3, b3, y3, 0);
  softmax_kernel<<<(n_nodes * 32 + 255) / 256, 256, 0, stream>>>(y3, n_nodes);
}